// GraphTransformerLayer_38491496907216
// MI455X (gfx1250) — compile-verified
//
#include <hip/hip_runtime.h>
#include <math.h>

#define NN 50000
#define EE 800000
#define DD 128
#define HH 8
#define DHH 16
#define SCAN_T 1024

typedef float v2f __attribute__((ext_vector_type(2)));
typedef float v8f __attribute__((ext_vector_type(8)));

// ---------------------------------------------------------------------------
// Kernel 1: fused QKV projection (emits v_wmma_f32_16x16x4_f32).
//   grid.x = NN/16 = 3125 blocks, 256 threads (8 waves).
// ---------------------------------------------------------------------------
__global__ __launch_bounds__(256)
void qkv_gemm_kernel(const float* __restrict__ x,
                     const float* __restrict__ Wq, const float* __restrict__ bq,
                     const float* __restrict__ Wk, const float* __restrict__ bk,
                     const float* __restrict__ Wv, const float* __restrict__ bv,
                     float* __restrict__ Q, float* __restrict__ K,
                     float* __restrict__ V)
{
    __shared__ float xs[16 * 132];

    const int tid  = threadIdx.x;
    const int wave = tid >> 5;      // 0..7 -> column tile
    const int lane = tid & 31;
    const int r0   = blockIdx.x * 16;

#pragma unroll
    for (int i = 0; i < 2; ++i) {
        int slot = tid + i * 256;                 // 0..511 float4 slots
        int row  = slot >> 5;
        int col4 = slot & 31;
        const float4 v = *(const float4*)(x + (size_t)(r0 + row) * DD + col4 * 4);
        float* d = &xs[row * 132 + col4 * 4];
        d[0] = v.x; d[1] = v.y; d[2] = v.z; d[3] = v.w;
    }
    __syncthreads();

    const int c0 = wave * 16;
    const int m  = lane & 15;
    const int h  = lane >> 4;

    const float* Ws[3] = {Wq, Wk, Wv};
    const float* bs[3] = {bq, bk, bv};
    float*       Os[3] = {Q,  K,  V };

#pragma unroll
    for (int mtx = 0; mtx < 3; ++mtx) {
        const float* __restrict__ W = Ws[mtx];
        v8f acc = {0.f, 0.f, 0.f, 0.f, 0.f, 0.f, 0.f, 0.f};
#pragma unroll 4
        for (int k0 = 0; k0 < DD; k0 += 4) {
            const int kk = k0 + 2 * h;
            v2f a, b;
            a.x = xs[m * 132 + kk];
            a.y = xs[m * 132 + kk + 1];
            b.x = W[(size_t)kk * DD + c0 + m];
            b.y = W[(size_t)(kk + 1) * DD + c0 + m];
            acc = __builtin_amdgcn_wmma_f32_16x16x4_f32(
                      false, a, false, b, (short)0, acc, false, false);
        }
        const float bias = bs[mtx][c0 + m];
        float* __restrict__ O = Os[mtx];
#pragma unroll
        for (int r = 0; r < 8; ++r) {
            O[(size_t)(r0 + r + 8 * h) * DD + c0 + m] = acc[r] + bias;
        }
    }
}

// ---------------------------------------------------------------------------
// Kernel 2: per-destination edge counts (counts pre-zeroed via memset).
// ---------------------------------------------------------------------------
__global__ __launch_bounds__(256)
void count_kernel(const int* __restrict__ dst, int* __restrict__ counts)
{
    const int e = blockIdx.x * blockDim.x + threadIdx.x;
    if (e < EE) atomicAdd(&counts[dst[e]], 1);
}

// ---------------------------------------------------------------------------
// Kernel 3: single-block exclusive prefix sum over counts[NN].
//   1024 threads x 49 sequential elems, Hillis-Steele over the partials.
//   Writes row_start[NN+1] and initializes cursor[] = row_start[].
// ---------------------------------------------------------------------------
__global__ __launch_bounds__(SCAN_T)
void scan_kernel(const int* __restrict__ counts, int* __restrict__ row_start,
                 int* __restrict__ cursor)
{
    __shared__ int partial[SCAN_T];
    const int t    = threadIdx.x;
    const int per  = (NN + SCAN_T - 1) / SCAN_T;   // 49
    const int base = t * per;

    int s = 0;
    for (int i = 0; i < per; ++i) {
        int idx = base + i;
        if (idx < NN) s += counts[idx];
    }
    partial[t] = s;
    __syncthreads();

    for (int off = 1; off < SCAN_T; off <<= 1) {
        int v = (t >= off) ? partial[t - off] : 0;
        __syncthreads();
        partial[t] += v;
        __syncthreads();
    }

    int run = (t > 0) ? partial[t - 1] : 0;        // exclusive offset of chunk
    for (int i = 0; i < per; ++i) {
        int idx = base + i;
        if (idx < NN) {
            row_start[idx] = run;
            cursor[idx]    = run;
            run += counts[idx];
        }
    }
    if (t == SCAN_T - 1) row_start[NN] = partial[SCAN_T - 1];  // == EE
}

// ---------------------------------------------------------------------------
// Kernel 4: scatter the SOURCE NODE ID of each edge into CSR-by-dst order.
//   (Storing src[e] directly removes one dependent load level from the
//    hot per-node loop.)
// ---------------------------------------------------------------------------
__global__ __launch_bounds__(256)
void scatter_kernel(const int* __restrict__ dst, const int* __restrict__ src,
                    int* __restrict__ cursor, int* __restrict__ srcs_csr)
{
    const int e = blockIdx.x * blockDim.x + threadIdx.x;
    if (e < EE) {
        const int pos = atomicAdd(&cursor[dst[e]], 1);
        srcs_csr[pos] = src[e];
    }
}

// ---------------------------------------------------------------------------
// Kernel 5: per-node attention, register accumulation, fused finalize.
//   One wave32 per node: lane l owns float4 chunk l (head = l>>2).
//   Index stream batched: each lane loads one src index per 32 edges
//   (coalesced), broadcast per edge via __shfl. No global atomics.
// ---------------------------------------------------------------------------
__global__ __launch_bounds__(256)
void node_attn_kernel(const float* __restrict__ x,
                      const float* __restrict__ Q, const float* __restrict__ K,
                      const float* __restrict__ V,
                      const int* __restrict__ srcs_csr,
                      const int* __restrict__ row_start,
                      float* __restrict__ out)
{
    const int lane = threadIdx.x & 31;
    const int n    = blockIdx.x * (blockDim.x >> 5) + (threadIdx.x >> 5);
    if (n >= NN) return;

    const int beg = row_start[n];
    const int end = row_start[n + 1];

    const float4 qv = ((const float4*)(Q + (size_t)n * DD))[lane];

    float4 acc = make_float4(0.f, 0.f, 0.f, 0.f);
    float  zs  = 0.f;

    for (int i0 = beg; i0 < end; i0 += 32) {
        const int rem  = end - i0;
        const int take = rem < 32 ? rem : 32;
        const int myS  = (lane < take) ? srcs_csr[i0 + lane] : 0;

        for (int j = 0; j < take; ++j) {
            const int s = __shfl(myS, j);     // broadcast edge's source node
            const float4 kv = ((const float4*)(K + (size_t)s * DD))[lane];
            const float4 vv = ((const float4*)(V + (size_t)s * DD))[lane];

            float p = kv.x * qv.x + kv.y * qv.y + kv.z * qv.z + kv.w * qv.w;
            p += __shfl_xor(p, 1);
            p += __shfl_xor(p, 2);            // 4 lanes of each head agree

            const float sc = __expf(fminf(fmaxf(p * 0.25f, -5.0f), 5.0f));
            acc.x += sc * vv.x;
            acc.y += sc * vv.y;
            acc.z += sc * vv.z;
            acc.w += sc * vv.w;
            zs    += sc;                      // same across the head's 4 lanes
        }
    }

    const float  inv = 1.0f / zs;             // every node has >=1 edge
    const float4 xv  = ((const float4*)(x + (size_t)n * DD))[lane];
    float4 o;
    o.x = xv.x + acc.x * inv;
    o.y = xv.y + acc.y * inv;
    o.z = xv.z + acc.z * inv;
    o.w = xv.w + acc.w * inv;
    ((float4*)(out + (size_t)n * DD))[lane] = o;
}

// ---------------------------------------------------------------------------
extern "C" void kernel_launch(void* const* d_in, const int* in_sizes, int n_in,
                              void* d_out, int out_size, void* d_ws, size_t ws_size,
                              hipStream_t stream)
{
    const float* x   = (const float*)d_in[0];
    const int*   src = (const int*)  d_in[1];
    const int*   dst = (const int*)  d_in[2];
    const float* Wq  = (const float*)d_in[3];
    const float* bq  = (const float*)d_in[4];
    const float* Wk  = (const float*)d_in[5];
    const float* bk  = (const float*)d_in[6];
    const float* Wv  = (const float*)d_in[7];
    const float* bv  = (const float*)d_in[8];
    float*       out = (float*)d_out;

    float* ws = (float*)d_ws;
    float* Q  = ws;
    float* K  = ws + 1 * (size_t)NN * DD;
    float* V  = ws + 2 * (size_t)NN * DD;
    int* counts    = (int*)(ws + 3 * (size_t)NN * DD);
    int* row_start = counts + NN;          // NN+1 entries
    int* cursor    = row_start + (NN + 1);
    int* srcs_csr  = cursor + NN;          // EE entries

    // zero only the 200 KB counts array
    hipMemsetAsync(counts, 0, (size_t)NN * sizeof(int), stream);

    qkv_gemm_kernel<<<NN / 16, 256, 0, stream>>>(x, Wq, bq, Wk, bk, Wv, bv,
                                                 Q, K, V);

    count_kernel<<<(EE + 255) / 256, 256, 0, stream>>>(dst, counts);

    scan_kernel<<<1, SCAN_T, 0, stream>>>(counts, row_start, cursor);

    scatter_kernel<<<(EE + 255) / 256, 256, 0, stream>>>(dst, src, cursor,
                                                         srcs_csr);

    node_attn_kernel<<<(NN + 7) / 8, 256, 0, stream>>>(x, Q, K, V, srcs_csr,
                                                       row_start, out);
}